// UPIFraudGNN_82033875353617
// MI455X (gfx1250) — compile-verified
//
#include <hip/hip_runtime.h>
#include <hip/hip_bf16.h>

#define NUv 100000
#define NMv 30000
#define NDv 50000
#define E_Tv 600000
#define E_Rv 600000
#define E_Uv 300000
#define E_Dv 300000
#define E_TMPv 600000
#define E_SIMv 400000

typedef float v2f __attribute__((ext_vector_type(2)));
typedef float v8f __attribute__((ext_vector_type(8)));

// ---------------------------------------------------------------------------
// fp32 WMMA tile MAC loop over one K-chunk.
// arow: per-lane row base (already includes the k-range start)
// Wm:   weight base offset to kbase row
// ---------------------------------------------------------------------------
__device__ __forceinline__ void wmma_chunk(const float* __restrict__ arow,
                                           const float* __restrict__ Wm,
                                           int Nc, int kcount, int hi, int ccl,
                                           bool cok, v8f& acc) {
  const v2f* ap = (const v2f*)(arow + (hi << 1));
  const float* bp = Wm + (long)(hi << 1) * Nc + ccl;
  long bstep = 4L * Nc;
#pragma unroll 4
  for (int k = 0; k < kcount; k += 4) {
    v2f a = *ap;
    ap += 2;
    float bx = bp[0];
    float by = bp[Nc];
    bp += bstep;
    v2f b;
    b.x = cok ? bx : 0.f;
    b.y = cok ? by : 0.f;
    acc = __builtin_amdgcn_wmma_f32_16x16x4_f32(false, a, false, b, (short)0, acc,
                                                false, false);
  }
}

// ---------------------------------------------------------------------------
// Generic fp32 WMMA GEMM: Y[M,Nc] (+)= act( A[M,K] @ W[K,Nc] + bias )
// One wave computes a 16x16 tile with chained V_WMMA_F32_16X16X4_F32.
// CONCAT: A row r = concat( A[m0[r]*64 ..], A[m1[r]*64 ..] ), K==128.
// flags: bit0 = accumulate into Y, bit1 = relu
// ---------------------------------------------------------------------------
template <bool CONCAT>
__global__ void gemm_tiles_k(const float* __restrict__ A, const int* __restrict__ m0,
                             const int* __restrict__ m1, const float* __restrict__ Wm,
                             const float* __restrict__ bias, float* __restrict__ Y,
                             int M, int K, int Nc, int ldy, int flags) {
  int lane = threadIdx.x & 31;
  int wid  = blockIdx.x * (blockDim.x >> 5) + (threadIdx.x >> 5);
  int tn = (Nc + 15) >> 4;
  int tm = (M + 15) >> 4;
  if (wid >= tm * tn) return;
  int tr = wid / tn, tc = wid - tr * tn;
  int row0 = tr << 4, col0 = tc << 4;
  int hi = lane >> 4, lo = lane & 15;
  int rr = min(row0 + lo, M - 1);

  int cc = col0 + lo;
  int ccl = min(cc, Nc - 1);
  bool cok = cc < Nc;

  v8f acc = {};
  if (CONCAT) {
    const float* arow0 = A + (long)m0[rr] * 64;
    const float* arow1 = A + (long)m1[rr] * 64;
    wmma_chunk(arow0, Wm, Nc, 64, hi, ccl, cok, acc);
    wmma_chunk(arow1, Wm + 64L * Nc, Nc, 64, hi, ccl, cok, acc);
  } else {
    const float* arow = A + (long)rr * K;
    wmma_chunk(arow, Wm, Nc, K, hi, ccl, cok, acc);
  }

  float bv = (bias != nullptr && cok) ? bias[cc] : 0.f;
#pragma unroll
  for (int i = 0; i < 8; i++) {
    int r2 = row0 + i + (hi << 3);
    if (r2 < M && cok) {
      float v = acc[i] + bv;
      if (flags & 2) v = fmaxf(v, 0.f);
      long idx = (long)r2 * ldy + cc;
      if (flags & 1) Y[idx] += v; else Y[idx] = v;
    }
  }
}

// ---------------------------------------------------------------------------
__global__ void fill_f32_k(float* p, float v, long n) {
  long t = blockIdx.x * (long)blockDim.x + threadIdx.x;
  if (t < n) p[t] = v;
}
__global__ void fill_u32_k(unsigned* p, unsigned v, long n) {
  long t = blockIdx.x * (long)blockDim.x + threadIdx.x;
  if (t < n) p[t] = v;
}

// SAGE mean-aggregation scatter: thread per (edge, channel)
__global__ void sage_scatter_k(const float* __restrict__ xs, const int* __restrict__ ei,
                               int E, float* __restrict__ agg, float* __restrict__ cnt) {
  long t = blockIdx.x * (long)blockDim.x + threadIdx.x;
  if (t >= (long)E * 64) return;
  int e = (int)(t >> 6), c = (int)(t & 63);
  int s = ei[e], d = ei[E + e];
  atomicAdd(&agg[(long)d * 64 + c], xs[(long)s * 64 + c]);
  if (c == 0) atomicAdd(&cnt[d], 1.0f);
}
__global__ void mean_div_k(float* agg, const float* cnt, long n64) {
  long t = blockIdx.x * (long)blockDim.x + threadIdx.x;
  if (t >= n64) return;
  agg[t] /= fmaxf(cnt[t >> 6], 1.0f);
}

// GAT: fold attention vectors: vsd[side][h][k] = sum_c w[k, h*64+c] * att[h,c]
__global__ void gat_vsd_k(const float* __restrict__ ws, const float* __restrict__ wd,
                          const float* __restrict__ as, const float* __restrict__ ad,
                          float* __restrict__ vsd) {
  int t = blockIdx.x * blockDim.x + threadIdx.x;
  if (t >= 512) return;
  int side = t >> 8, h = (t >> 6) & 3, k = t & 63;
  const float* w = side ? wd : ws;
  const float* a = side ? ad : as;
  float sum = 0.f;
  for (int c = 0; c < 64; c++) sum += w[k * 256 + h * 64 + c] * a[h * 64 + c];
  vsd[t] = sum;
}
// a_s / a_d = x @ vsd
__global__ void gat_asd_k(const float* __restrict__ xu, const float* __restrict__ vsd,
                          float* __restrict__ as_, float* __restrict__ ad_, int N) {
  long t = blockIdx.x * (long)blockDim.x + threadIdx.x;
  if (t >= (long)N * 8) return;
  int n = (int)(t >> 3), q = (int)(t & 7);
  int side = q >> 2, h = q & 3;
  const float* v = vsd + side * 256 + h * 64;
  const float* x = xu + (long)n * 64;
  float s = 0.f;
  for (int k = 0; k < 64; k++) s += x[k] * v[k];
  (side ? ad_ : as_)[(long)n * 4 + h] = s;
}
__device__ inline unsigned encf(float f) {
  int i = __float_as_int(f);
  return (i >= 0) ? ((unsigned)i | 0x80000000u) : ~(unsigned)i;
}
__device__ inline float decf(unsigned u) {
  int i = (u & 0x80000000u) ? (int)(u & 0x7FFFFFFFu) : (int)(~u);
  return __int_as_float(i);
}
__global__ void gat_max_k(const int* __restrict__ ei, int E, int N,
                          const float* __restrict__ as_, const float* __restrict__ ad_,
                          unsigned* __restrict__ m) {
  long t = blockIdx.x * (long)blockDim.x + threadIdx.x;
  if (t >= (long)(E + N) * 4) return;
  int e = (int)(t >> 2), h = (int)(t & 3);
  int s, d;
  if (e < E) { s = ei[e]; d = ei[E + e]; } else { s = e - E; d = s; }
  float v = as_[(long)s * 4 + h] + ad_[(long)d * 4 + h];
  v = v > 0.f ? v : 0.2f * v;
  atomicMax(&m[(long)d * 4 + h], encf(v));
}
__global__ void gat_exp_k(const int* __restrict__ ei, int E, int N,
                          const float* __restrict__ as_, const float* __restrict__ ad_,
                          const unsigned* __restrict__ m, float* __restrict__ exb,
                          float* __restrict__ den) {
  long t = blockIdx.x * (long)blockDim.x + threadIdx.x;
  if (t >= (long)(E + N) * 4) return;
  int e = (int)(t >> 2), h = (int)(t & 3);
  int s, d;
  if (e < E) { s = ei[e]; d = ei[E + e]; } else { s = e - E; d = s; }
  float v = as_[(long)s * 4 + h] + ad_[(long)d * 4 + h];
  v = v > 0.f ? v : 0.2f * v;
  float ex = expf(v - decf(m[(long)d * 4 + h]));
  exb[t] = ex;
  atomicAdd(&den[(long)d * 4 + h], ex);
}
// head-mean folded into scatter: nu[d,c] += 0.25 * sum_h alpha * hs[s, h, c]
__global__ void gat_scatter_k(const int* __restrict__ ei, int E, int N,
                              const float* __restrict__ hs, const float* __restrict__ exb,
                              const float* __restrict__ den, float* __restrict__ nu) {
  long t = blockIdx.x * (long)blockDim.x + threadIdx.x;
  if (t >= (long)(E + N) * 64) return;
  int e = (int)(t >> 6), c = (int)(t & 63);
  int s, d;
  if (e < E) { s = ei[e]; d = ei[E + e]; } else { s = e - E; d = s; }
  float val = 0.f;
#pragma unroll
  for (int h = 0; h < 4; h++) {
    float al = exb[(long)e * 4 + h] / (den[(long)d * 4 + h] + 1e-16f);
    val += al * hs[(long)s * 256 + h * 64 + c];
  }
  atomicAdd(&nu[(long)d * 64 + c], 0.25f * val);
}

// GCN
__global__ void gcn_deg_k(const int* __restrict__ ei, int E, float* __restrict__ cnt) {
  long t = blockIdx.x * (long)blockDim.x + threadIdx.x;
  if (t >= E) return;
  atomicAdd(&cnt[ei[E + t]], 1.0f);
}
__global__ void rsqrt_k(float* p, long n) {
  long t = blockIdx.x * (long)blockDim.x + threadIdx.x;
  if (t < n) p[t] = rsqrtf(fmaxf(p[t], 1e-12f));
}
__global__ void gcn_scatter_k(const int* __restrict__ ei, int E, int N,
                              const float* __restrict__ h, const float* __restrict__ dis,
                              float* __restrict__ nu) {
  long t = blockIdx.x * (long)blockDim.x + threadIdx.x;
  if (t >= (long)(E + N) * 64) return;
  int e = (int)(t >> 6), c = (int)(t & 63);
  int s, d;
  if (e < E) { s = ei[e]; d = ei[E + e]; } else { s = e - E; d = s; }
  atomicAdd(&nu[(long)d * 64 + c], h[(long)s * 64 + c] * dis[s] * dis[d]);
}
__global__ void add_bias2_k(float* x, const float* b1, const float* b2, long n64) {
  long t = blockIdx.x * (long)blockDim.x + threadIdx.x;
  if (t >= n64) return;
  x[t] += b1[t & 63] + b2[t & 63];
}

// BatchNorm (biased var) + ReLU
__global__ void bn_stats_k(const float* __restrict__ x, int N, float* __restrict__ stat) {
  int c = threadIdx.x & 63;
  int r0 = blockIdx.x * 4 + (threadIdx.x >> 6);
  int stride = gridDim.x * 4;
  float s = 0.f, ss = 0.f;
  for (int r = r0; r < N; r += stride) {
    float v = x[(long)r * 64 + c];
    s += v; ss += v * v;
  }
  atomicAdd(&stat[c], s);
  atomicAdd(&stat[64 + c], ss);
}
__global__ void bn_apply_k(const float* __restrict__ x, float* __restrict__ y, int N,
                           const float* __restrict__ stat, const float* __restrict__ gamma,
                           const float* __restrict__ beta) {
  long t = blockIdx.x * (long)blockDim.x + threadIdx.x;
  if (t >= (long)N * 64) return;
  int c = (int)(t & 63);
  float mu = stat[c] / (float)N;
  float var = fmaxf(stat[64 + c] / (float)N - mu * mu, 0.f);
  float v = (x[t] - mu) * rsqrtf(var + 1e-5f) * gamma[c] + beta[c];
  y[t] = fmaxf(v, 0.f);
}

// graph mean-pool
__global__ void pool_scatter_k(const float* __restrict__ x, const int* __restrict__ b,
                               int N, float* __restrict__ g, float* __restrict__ cnt,
                               int base) {
  long t = blockIdx.x * (long)blockDim.x + threadIdx.x;
  if (t >= (long)N * 64) return;
  int n = (int)(t >> 6), c = (int)(t & 63);
  int gr = b[n];
  atomicAdd(&g[(long)gr * 192 + base + c], x[t]);
  if (c == 0) atomicAdd(&cnt[gr], 1.0f);
}
__global__ void pool_div_k(float* g, const float* gcnt) {
  int t = blockIdx.x * blockDim.x + threadIdx.x;
  if (t >= 64 * 192) return;
  int gr = t / 192, c = t - gr * 192, type = c >> 6;
  g[t] /= fmaxf(gcnt[type * 64 + gr], 1.0f);
}

// ---------------------------------------------------------------------------
static inline unsigned cdiv_l(long a, long b) { return (unsigned)((a + b - 1) / b); }

static inline void gemm(hipStream_t st, const float* A, const int* m0, const int* m1,
                        const float* Wm, const float* bias, float* Y,
                        int M, int K, int Nc, int ldy, int flags) {
  long tiles = (long)((M + 15) / 16) * ((Nc + 15) / 16);
  if (m0)
    gemm_tiles_k<true><<<dim3(cdiv_l(tiles, 8)), dim3(256), 0, st>>>(
        A, m0, m1, Wm, bias, Y, M, K, Nc, ldy, flags);
  else
    gemm_tiles_k<false><<<dim3(cdiv_l(tiles, 8)), dim3(256), 0, st>>>(
        A, m0, m1, Wm, bias, Y, M, K, Nc, ldy, flags);
}
static inline void fillf(hipStream_t st, float* p, float v, long n) {
  fill_f32_k<<<dim3(cdiv_l(n, 256)), dim3(256), 0, st>>>(p, v, n);
}
static inline void fillu(hipStream_t st, unsigned* p, unsigned v, long n) {
  fill_u32_k<<<dim3(cdiv_l(n, 256)), dim3(256), 0, st>>>(p, v, n);
}
#define L1D(kern, n, ...) kern<<<dim3(cdiv_l((long)(n), 256)), dim3(256), 0, stream>>>(__VA_ARGS__)

extern "C" void kernel_launch(void* const* d_in, const int* in_sizes, int n_in,
                              void* d_out, int out_size, void* d_ws, size_t ws_size,
                              hipStream_t stream) {
  (void)in_sizes; (void)n_in; (void)out_size; (void)ws_size;
  const float* x_user     = (const float*)d_in[0];
  const float* x_merchant = (const float*)d_in[1];
  const float* x_device   = (const float*)d_in[2];
  const int* ei_t  = (const int*)d_in[3];
  const int* ei_r  = (const int*)d_in[4];
  const int* ei_u  = (const int*)d_in[5];
  const int* ei_d  = (const int*)d_in[6];
  const int* ei_tm = (const int*)d_in[7];
  const int* ei_s  = (const int*)d_in[8];
  const int* b_u   = (const int*)d_in[9];
  const int* b_m   = (const int*)d_in[10];
  const int* b_d   = (const int*)d_in[11];
  // params flattened in JAX pytree order (dict keys sorted)
  auto F = [&](int i) { return (const float*)d_in[i]; };

  // workspace carve (floats)
  float* base = (float*)d_ws;
  size_t o = 0;
  auto AL = [&](size_t n) { float* p = base + o; o += n; return p; };
  float* xu   = AL((size_t)NUv * 64);
  float* xm   = AL((size_t)NMv * 64);
  float* xd   = AL((size_t)NDv * 64);
  float* nu   = AL((size_t)NUv * 64);
  float* nm   = AL((size_t)NMv * 64);
  float* nd   = AL((size_t)NDv * 64);
  float* mean = AL((size_t)NUv * 64);   // sage means / gcn hidden
  float* cnt  = AL((size_t)NUv);        // sage counts / gcn degree
  float* hs   = AL((size_t)NUv * 256);  // GAT per-head features
  float* exb  = AL((size_t)(E_TMPv + NUv) * 4);
  float* as_  = AL((size_t)NUv * 4);
  float* ad_  = AL((size_t)NUv * 4);
  unsigned* mrow = (unsigned*)AL((size_t)NUv * 4);
  float* den  = AL((size_t)NUv * 4);
  float* vsd  = AL(512);
  float* stat = AL(128);
  float* g    = AL(64 * 192);
  float* gcnt = AL(192);
  float* uh   = AL((size_t)NUv * 32);
  float* mh   = AL((size_t)NMv * 32);
  const int ECH = 120000;
  float* eh1  = AL((size_t)ECH * 64);
  float* eh2  = AL((size_t)ECH * 32);
  float* gh1  = AL(64 * 64);
  float* gh2  = AL(64 * 32);
  float* out  = (float*)d_out;

  // ---- projections: proj.{device,merchant,user} at 107..112 ----
  gemm(stream, x_user,     nullptr, nullptr, F(112), F(111), xu, NUv, 32, 64, 64, 0);
  gemm(stream, x_merchant, nullptr, nullptr, F(110), F(109), xm, NMv, 24, 64, 64, 0);
  gemm(stream, x_device,   nullptr, nullptr, F(108), F(107), xd, NDv, 16, 64, 64, 0);

  for (int l = 0; l < 3; l++) {
    int LB = 32 + 25 * l;
    auto P = [&](int j) { return (const float*)d_in[LB + j]; };
    // j: 0..5 bn(dev.b,dev.g,mer.b,mer.g,usr.b,usr.g) 6 att_dst 7 att_src 8 gat.bias
    //    9 w_dst 10 w_src 11 gcn.bias 12 gcn.w
    //    13..15 sage_du(lb,lw,rw) 16..18 sage_mu 19..21 sage_tm 22..24 sage_ud

    // new_m = sage_tm(xu -> merchants)
    fillf(stream, mean, 0.f, (long)NMv * 64); fillf(stream, cnt, 0.f, NMv);
    L1D(sage_scatter_k, (long)E_Tv * 64, xu, ei_t, E_Tv, mean, cnt);
    L1D(mean_div_k, (long)NMv * 64, mean, cnt, (long)NMv * 64);
    gemm(stream, mean, 0, 0, P(20), P(19), nm, NMv, 64, 64, 64, 0);
    gemm(stream, xm,   0, 0, P(21), nullptr, nm, NMv, 64, 64, 64, 1);

    // new_d = sage_ud(xu -> devices)
    fillf(stream, mean, 0.f, (long)NDv * 64); fillf(stream, cnt, 0.f, NDv);
    L1D(sage_scatter_k, (long)E_Uv * 64, xu, ei_u, E_Uv, mean, cnt);
    L1D(mean_div_k, (long)NDv * 64, mean, cnt, (long)NDv * 64);
    gemm(stream, mean, 0, 0, P(23), P(22), nd, NDv, 64, 64, 64, 0);
    gemm(stream, xd,   0, 0, P(24), nullptr, nd, NDv, 64, 64, 64, 1);

    // new_u = sage_mu(xm -> users)
    fillf(stream, mean, 0.f, (long)NUv * 64); fillf(stream, cnt, 0.f, NUv);
    L1D(sage_scatter_k, (long)E_Rv * 64, xm, ei_r, E_Rv, mean, cnt);
    L1D(mean_div_k, (long)NUv * 64, mean, cnt, (long)NUv * 64);
    gemm(stream, mean, 0, 0, P(17), P(16), nu, NUv, 64, 64, 64, 0);
    gemm(stream, xu,   0, 0, P(18), nullptr, nu, NUv, 64, 64, 64, 1);
    //        + sage_du(xd -> users)
    fillf(stream, mean, 0.f, (long)NUv * 64); fillf(stream, cnt, 0.f, NUv);
    L1D(sage_scatter_k, (long)E_Dv * 64, xd, ei_d, E_Dv, mean, cnt);
    L1D(mean_div_k, (long)NUv * 64, mean, cnt, (long)NUv * 64);
    gemm(stream, mean, 0, 0, P(14), P(13), nu, NUv, 64, 64, 64, 1);
    gemm(stream, xu,   0, 0, P(15), nullptr, nu, NUv, 64, 64, 64, 1);

    //        + GAT(xu, ei_temporal)
    gat_vsd_k<<<2, 256, 0, stream>>>(P(10), P(9), P(7), P(6), vsd);
    L1D(gat_asd_k, (long)NUv * 8, xu, vsd, as_, ad_, NUv);
    gemm(stream, xu, 0, 0, P(10), nullptr, hs, NUv, 64, 256, 256, 0);
    fillu(stream, mrow, 0x007FFFFFu, (long)NUv * 4);  // encf(-inf)
    fillf(stream, den, 0.f, (long)NUv * 4);
    L1D(gat_max_k, (long)(E_TMPv + NUv) * 4, ei_tm, E_TMPv, NUv, as_, ad_, mrow);
    L1D(gat_exp_k, (long)(E_TMPv + NUv) * 4, ei_tm, E_TMPv, NUv, as_, ad_, mrow, exb, den);
    L1D(gat_scatter_k, (long)(E_TMPv + NUv) * 64, ei_tm, E_TMPv, NUv, hs, exb, den, nu);

    //        + GCN(xu, ei_similar)
    fillf(stream, cnt, 1.f, NUv);                      // self-loop degree
    L1D(gcn_deg_k, E_SIMv, ei_s, E_SIMv, cnt);
    L1D(rsqrt_k, NUv, cnt, (long)NUv);
    gemm(stream, xu, 0, 0, P(12), nullptr, mean, NUv, 64, 64, 64, 0);
    L1D(gcn_scatter_k, (long)(E_SIMv + NUv) * 64, ei_s, E_SIMv, NUv, mean, cnt, nu);
    L1D(add_bias2_k, (long)NUv * 64, nu, P(8), P(11), (long)NUv * 64);

    // BN + ReLU -> next-layer features
    fillf(stream, stat, 0.f, 128);
    bn_stats_k<<<256, 256, 0, stream>>>(nu, NUv, stat);
    L1D(bn_apply_k, (long)NUv * 64, nu, xu, NUv, stat, P(5), P(4));
    fillf(stream, stat, 0.f, 128);
    bn_stats_k<<<256, 256, 0, stream>>>(nm, NMv, stat);
    L1D(bn_apply_k, (long)NMv * 64, nm, xm, NMv, stat, P(3), P(2));
    fillf(stream, stat, 0.f, 128);
    bn_stats_k<<<256, 256, 0, stream>>>(nd, NDv, stat);
    L1D(bn_apply_k, (long)NDv * 64, nd, xd, NDv, stat, P(1), P(0));
  }

  // ---- detection heads ----
  // user_cls (28..31), merchant_cls (24..27)
  gemm(stream, xu, 0, 0, F(29), F(28), uh, NUv, 64, 32, 32, 2);
  gemm(stream, uh, 0, 0, F(31), F(30), out + 0, NUv, 32, 2, 2, 0);
  gemm(stream, xm, 0, 0, F(25), F(24), mh, NMv, 64, 32, 32, 2);
  gemm(stream, mh, 0, 0, F(27), F(26), out + 200000, NMv, 32, 2, 2, 0);

  // edge_cls (12..17): fused concat-gather GEMM, chunked
  for (int s0 = 0; s0 < E_TMPv; s0 += ECH) {
    int len = (E_TMPv - s0) < ECH ? (E_TMPv - s0) : ECH;
    gemm(stream, xu, ei_tm + s0, ei_tm + E_TMPv + s0, F(13), F(12), eh1, len, 128, 64, 64, 2);
    gemm(stream, eh1, 0, 0, F(15), F(14), eh2, len, 64, 32, 32, 2);
    gemm(stream, eh2, 0, 0, F(17), F(16), out + 260000 + s0, len, 32, 1, 1, 0);
  }

  // graph_cls (18..23): mean-pool then MLP
  fillf(stream, g, 0.f, 64 * 192); fillf(stream, gcnt, 0.f, 192);
  L1D(pool_scatter_k, (long)NUv * 64, xu, b_u, NUv, g, gcnt + 0, 0);
  L1D(pool_scatter_k, (long)NMv * 64, xm, b_m, NMv, g, gcnt + 64, 64);
  L1D(pool_scatter_k, (long)NDv * 64, xd, b_d, NDv, g, gcnt + 128, 128);
  pool_div_k<<<48, 256, 0, stream>>>(g, gcnt);
  gemm(stream, g,   0, 0, F(19), F(18), gh1, 64, 192, 64, 64, 2);
  gemm(stream, gh1, 0, 0, F(21), F(20), gh2, 64, 64, 32, 32, 2);
  gemm(stream, gh2, 0, 0, F(23), F(22), out + 860000, 64, 32, 1, 1, 0);
}